// CDE_BCR_46299747451577
// MI455X (gfx1250) — compile-verified
//
#include <hip/hip_runtime.h>

// CDNA5 / gfx1250: wave32, WMMA f32 16x16x4.
typedef __attribute__((ext_vector_type(2))) float v2f;
typedef __attribute__((ext_vector_type(8))) float v8f;

#define BB 4
#define DD 32
#define KK 8
#define CI 2
#define CO 2
#define LL 4096
#define NB 15
#define SS 8
#define PP 7
#define SEGLEN 499

__device__ __forceinline__ int seg_of(int l) {
    int s = l / SEGLEN;
    return s > (SS - 1) ? (SS - 1) : s;
}

// Edge-tile x load: clamped in-bounds address, select 0 when logically OOB.
__device__ __forceinline__ float ldx(const float* __restrict__ p, int u) {
    int uc = u < 0 ? 0 : (u > (LL - 1) ? (LL - 1) : u);
    float v = p[uc];
    return (u >= 0 && u < LL) ? v : 0.0f;
}

// Load per-segment weights into WMMA B-operand layout.
// B (4x16 per step): lane n<16 half h holds rows kk0+2h+{0,1}; columns n>=CO and
// pad rows f>=NB are zero (which also makes A's pad-slot contents irrelevant).
__device__ __forceinline__ void load_bw(const float* __restrict__ wb, int s,
                                        int row, int ro, int half, v2f bw[8]) {
#pragma unroll
    for (int st = 0; st < 8; ++st) {
#pragma unroll
        for (int j = 0; j < 2; ++j) {
            const int kk = st * 4 + half * 2 + j;
            const int ci = kk >> 4;
            const int f  = kk & 15;
            const int fc = f > (NB - 1) ? (NB - 1) : f;
            float v = wb[((ro * CI + ci) * SS + s) * NB + fc];
            bw[st][j] = (row < CO && f < NB) ? v : 0.0f;
        }
    }
}

__global__ __launch_bounds__(256) void conv_seg_wmma(
    const float* __restrict__ x,      // (B,D,K,CI,L)
    const float* __restrict__ w,      // (D,K,CO,CI,S,1,NB)
    const float* __restrict__ bias,   // (D,K,CO,S,1)
    float* __restrict__ out)          // (B,D,K,CO,L)
{
    const int combo = blockIdx.x;              // [0, B*D*K)
    const int b   = combo / (DD * KK);
    const int rem = combo % (DD * KK);
    const int d   = rem / KK;
    const int k   = rem % KK;

    const float* __restrict__ xb = x    + (size_t)(((b * DD + d) * KK + k)) * (CI * LL);
    const float* __restrict__ wb = w    + (size_t)(((d * KK + k) * CO))     * (CI * SS * NB);
    const float* __restrict__ bb = bias + (size_t)(((d * KK + k) * CO))     * SS;
    float*       __restrict__ ob = out  + (size_t)((((b * DD + d) * KK + k) * CO)) * LL;

    const int tid  = threadIdx.x;
    const int lane = tid & 31;
    const int half = lane >> 4;    // lane half: selects K-slot pair
    const int row  = lane & 15;    // M row for A, N column for B (N=o, only o<2 used)
    const int ro   = row < (CO - 1) ? row : (CO - 1);   // clamped o for safe addressing

    // Wave-uniform base of this wave's 512-element l range (scalar for uniform CF).
    const int l0_base = __builtin_amdgcn_readfirstlane((tid >> 5) * (LL / 8));

    v2f bw[8];                 // B operand (weights) for current segment
    float biasv = 0.0f;
    int cur_seg = -1;

    for (int t = 0; t < 32; ++t) {
        const int l0 = l0_base + t * 16;                // scalar
        const int s0 = seg_of(l0);                      // scalar
        const int s1 = seg_of(l0 + 15);                 // scalar

        if (s0 != cur_seg) {                            // scalar branch
            load_bw(wb, s0, row, ro, half, bw);
            biasv = (row < CO) ? bb[ro * SS + s0] : 0.0f;
            cur_seg = s0;
        }

        if (t + 1 < 32) {   // prefetch next tile's x window
            __builtin_prefetch(xb + (l0 + 16), 0, 1);
            __builtin_prefetch(xb + LL + (l0 + 16), 0, 1);
        }

        const bool edge = (l0 == 0) | (l0 == LL - 16);  // scalar, rare
        v8f c = {0.f, 0.f, 0.f, 0.f, 0.f, 0.f, 0.f, 0.f};

        if (!edge) {
            if (s1 == s0) {
                // ---- hot path: unconditional loads, 8 chained WMMAs ----
#pragma unroll
                for (int st = 0; st < 8; ++st) {
                    const int kk0 = st * 4 + half * 2;
                    const int ci  = kk0 >> 4;
                    const int f0  = kk0 & 15;
                    const float* p = xb + ci * LL + (l0 + row + f0 - PP);
                    v2f a; a[0] = p[0]; a[1] = p[1];
                    c = __builtin_amdgcn_wmma_f32_16x16x4_f32(
                            false, a, false, bw[st], (short)0, c, false, false);
                }
                if (row < CO) {
                    float* dst = ob + row * LL + (l0 + half * 8);
                    *reinterpret_cast<float4*>(dst) =
                        make_float4(c[0] + biasv, c[1] + biasv, c[2] + biasv, c[3] + biasv);
                    *reinterpret_cast<float4*>(dst + 4) =
                        make_float4(c[4] + biasv, c[5] + biasv, c[6] + biasv, c[7] + biasv);
                }
            } else {
                // ---- straddle path (<=2 tiles per wave): dual accumulators ----
                v2f bw2[8];
                load_bw(wb, s1, row, ro, half, bw2);
                const float biasv2 = (row < CO) ? bb[ro * SS + s1] : 0.0f;

                v8f c2 = {0.f, 0.f, 0.f, 0.f, 0.f, 0.f, 0.f, 0.f};
#pragma unroll
                for (int st = 0; st < 8; ++st) {
                    const int kk0 = st * 4 + half * 2;
                    const int ci  = kk0 >> 4;
                    const int f0  = kk0 & 15;
                    const float* p = xb + ci * LL + (l0 + row + f0 - PP);
                    v2f a; a[0] = p[0]; a[1] = p[1];
                    c  = __builtin_amdgcn_wmma_f32_16x16x4_f32(
                             false, a, false, bw[st],  (short)0, c,  false, false);
                    c2 = __builtin_amdgcn_wmma_f32_16x16x4_f32(
                             false, a, false, bw2[st], (short)0, c2, false, false);
                }
                if (row < CO) {
                    const int bpos = (s0 + 1) * SEGLEN;
                    float vals[8];
#pragma unroll
                    for (int r = 0; r < 8; ++r) {
                        const int l = l0 + r + half * 8;   // D: M = r + 8*half
                        vals[r] = (l >= bpos) ? (c2[r] + biasv2) : (c[r] + biasv);
                    }
                    float* dst = ob + row * LL + (l0 + half * 8);
                    *reinterpret_cast<float4*>(dst)     = make_float4(vals[0], vals[1], vals[2], vals[3]);
                    *reinterpret_cast<float4*>(dst + 4) = make_float4(vals[4], vals[5], vals[6], vals[7]);
                }
            }
        } else {
            // ---- edge path (first/last tile only): bounds-checked loads ----
#pragma unroll
            for (int st = 0; st < 8; ++st) {
                const int kk0 = st * 4 + half * 2;
                const int ci  = kk0 >> 4;
                const int f0  = kk0 & 15;
                const int u0  = l0 + row + f0 - PP;
                v2f a;
                a[0] = ldx(xb + ci * LL, u0);
                a[1] = ldx(xb + ci * LL, u0 + 1);
                c = __builtin_amdgcn_wmma_f32_16x16x4_f32(
                        false, a, false, bw[st], (short)0, c, false, false);
            }
            if (row < CO) {
                float* dst = ob + row * LL + (l0 + half * 8);
                *reinterpret_cast<float4*>(dst) =
                    make_float4(c[0] + biasv, c[1] + biasv, c[2] + biasv, c[3] + biasv);
                *reinterpret_cast<float4*>(dst + 4) =
                    make_float4(c[4] + biasv, c[5] + biasv, c[6] + biasv, c[7] + biasv);
            }
        }
    }
}

extern "C" void kernel_launch(void* const* d_in, const int* in_sizes, int n_in,
                              void* d_out, int out_size, void* d_ws, size_t ws_size,
                              hipStream_t stream) {
    (void)in_sizes; (void)n_in; (void)d_ws; (void)ws_size; (void)out_size;
    const float* x    = (const float*)d_in[0];
    const float* wgt  = (const float*)d_in[1];
    const float* bias = (const float*)d_in[2];
    float* out = (float*)d_out;

    dim3 grid(BB * DD * KK);   // 1024 workgroups, one (b,d,k) each
    dim3 block(256);           // 8 wave32s, each owns 512 l-positions
    hipLaunchKernelGGL(conv_seg_wmma, grid, block, 0, stream, x, wgt, bias, out);
}